// ChannelGroupedMemory_34368328302980
// MI455X (gfx1250) — compile-verified
//
#include <hip/hip_runtime.h>
#include <hip/hip_bf16.h>
#include <math.h>

#define G_TOT        4096
#define D_DIM        512
#define M_SL         32
#define TOPK         3
#define BLOCK        256
#define GRPS_PER_BLK 4
#define NBLK         (G_TOT / GRPS_PER_BLK)

typedef __attribute__((ext_vector_type(2))) float v2f;
typedef __attribute__((ext_vector_type(8))) float v8f;

// exact pointee types expected by the async-to-LDS builtins (per hipcc diag)
typedef int v4i_vs __attribute__((vector_size(16)));
typedef int v2i_vs __attribute__((vector_size(8)));
typedef __attribute__((address_space(1))) v4i_vs* g4p;
typedef __attribute__((address_space(3))) v4i_vs* l4p;
typedef __attribute__((address_space(1))) v2i_vs* g2p;
typedef __attribute__((address_space(3))) v2i_vs* l2p;

#if __has_builtin(__builtin_amdgcn_global_load_async_to_lds_b128)
#define HAVE_ASYNC_B128 1
#endif
#if __has_builtin(__builtin_amdgcn_global_load_async_to_lds_b64)
#define HAVE_ASYNC_B64 1
#endif
#if __has_builtin(__builtin_amdgcn_wmma_f32_16x16x4_f32)
#define HAVE_WMMA_F32 1
#endif

// --- number of async ops issued per thread per group prefetch --------------
#if defined(HAVE_ASYNC_B128) && defined(HAVE_ASYNC_B64)
#define PF_OPS 17
#elif defined(HAVE_ASYNC_B128)
#define PF_OPS 16
#else
#define PF_OPS 0
#endif

__device__ __forceinline__ void async_copy_b128(const float* g, float* l) {
#if defined(HAVE_ASYNC_B128)
  __builtin_amdgcn_global_load_async_to_lds_b128((g4p)(void*)g, (l4p)(void*)l,
                                                 0, 0);
#else
  *(float4*)l = *(const float4*)g;
#endif
}

__device__ __forceinline__ void async_copy_b64(const float* g, float* l) {
#if defined(HAVE_ASYNC_B64)
  __builtin_amdgcn_global_load_async_to_lds_b64((g2p)(void*)g, (l2p)(void*)l,
                                                0, 0);
#else
  *(float2*)l = *(const float2*)g;
#endif
}

template <int N>
__device__ __forceinline__ void wait_async() {
#if defined(HAVE_ASYNC_B128)
#if __has_builtin(__builtin_amdgcn_s_wait_asynccnt)
  __builtin_amdgcn_s_wait_asynccnt(N);
#else
  asm volatile("s_wait_asynccnt %0" ::"n"(N) : "memory");
#endif
#endif
}

// stage one group's W (64KB) + q (2KB) into LDS; PF_OPS async ops per thread
__device__ __forceinline__ void prefetch_group(const float* __restrict__ Wg,
                                               const float* __restrict__ qg,
                                               float* wb, float* qb, int tid) {
#pragma unroll
  for (int j = 0; j < 16; ++j) {
    const int f4 = j * BLOCK + tid;           // float4 index within 16384 floats
    async_copy_b128(Wg + f4 * 4, wb + f4 * 4);
  }
  async_copy_b64(qg + tid * 2, qb + tid * 2); // 256 threads * 2 floats = 512
}

extern "C" __global__ void __launch_bounds__(BLOCK)
cgm_retrieve_kernel(const float* __restrict__ q,           // [G,D]
                    const float* __restrict__ W,           // [G,D,M]
                    const float* __restrict__ temp_logit,  // [G]
                    const float* __restrict__ shift_p,     // [1]
                    float* __restrict__ out) {
  __shared__ float wbuf[2][D_DIM * M_SL];   // 128 KB double buffer
  __shared__ float qbuf[2][D_DIM];          // 4 KB
  __shared__ float att_part[8][2][M_SL];    // per-(wave,half) partials
  __shared__ float att_final[M_SL];

  const int tid  = threadIdx.x;
  const int wv   = tid >> 5;   // wave 0..7
  const int ln   = tid & 31;
  const int half = ln >> 4;    // lane half (K-offset 0 or 2)
  const int col  = ln & 15;    // N / slot column
  const int g0   = blockIdx.x * GRPS_PER_BLK;
  const float shift = shift_p[0];

  float* out_ret = out;
  float* out_idx = out + (size_t)G_TOT * D_DIM;
  float* out_val = out_idx + (size_t)G_TOT * TOPK;

  // prologue: prefetch first group into buffer 0
  prefetch_group(W + (size_t)g0 * D_DIM * M_SL, q + (size_t)g0 * D_DIM,
                 wbuf[0], qbuf[0], tid);

  for (int it = 0; it < GRPS_PER_BLK; ++it) {
    const int g   = g0 + it;
    const int cur = it & 1;
    const int nxt = cur ^ 1;

    if (it + 1 < GRPS_PER_BLK) {
      const int gn = g + 1;
      prefetch_group(W + (size_t)gn * D_DIM * M_SL, q + (size_t)gn * D_DIM,
                     wbuf[nxt], qbuf[nxt], tid);
      wait_async<PF_OPS>();   // leave only the just-issued prefetch in flight
    } else {
      wait_async<0>();
    }
    __syncthreads();          // everyone's copies for buf[cur] are visible

    const float* wb = wbuf[cur];
    const float* qb = qbuf[cur];

    // ---- att = q . W via V_WMMA_F32_16X16X4_F32 ------------------------
    // A (16x4): q segment replicated over M rows. B (4x16): W rows x slots.
    // C accumulates att[slot] replicated over M. Two chains: slots 0-15,16-31.
    // Wave wv owns K-chunks [wv*16, wv*16+16) of the 128 chunks (K=4 each).
#if defined(HAVE_WMMA_F32)
    v8f c0 = {};
    v8f c1 = {};
#else
    float f0 = 0.f, f1 = 0.f;
#endif
#pragma unroll
    for (int cc = 0; cc < 16; ++cc) {
      const int dk = (wv * 16 + cc) * 4 + 2 * half;  // this lane's first d
      v2f a;
      a[0] = qb[dk];
      a[1] = qb[dk + 1];
      v2f b0;
      b0[0] = wb[dk * M_SL + col];
      b0[1] = wb[(dk + 1) * M_SL + col];
      v2f b1;
      b1[0] = wb[dk * M_SL + 16 + col];
      b1[1] = wb[(dk + 1) * M_SL + 16 + col];
#if defined(HAVE_WMMA_F32)
      c0 = __builtin_amdgcn_wmma_f32_16x16x4_f32(false, a, false, b0,
                                                 (short)0, c0, false, false);
      c1 = __builtin_amdgcn_wmma_f32_16x16x4_f32(false, a, false, b1,
                                                 (short)0, c1, false, false);
#else
      f0 += a[0] * b0[0] + a[1] * b0[1];
      f1 += a[0] * b1[0] + a[1] * b1[1];
#endif
    }
#if defined(HAVE_WMMA_F32)
    // every lane of C holds att[slot of its column]; keep half==0 copy only
    const float p0 = half ? 0.f : c0[0];
    const float p1 = half ? 0.f : c1[0];
#else
    const float p0 = f0, p1 = f1;   // per-lane partial (halves hold distinct K)
#endif
    att_part[wv][half][col]      = p0;
    att_part[wv][half][16 + col] = p1;
    __syncthreads();

    // fixed-order deterministic reduction across the 8 waves
    if (tid < M_SL) {
      float s = 0.f;
#pragma unroll
      for (int ww = 0; ww < 8; ++ww) {
        s += att_part[ww][0][tid];
        s += att_part[ww][1][tid];
      }
      att_final[tid] = s;
    }
    __syncthreads();

    // ---- softmax + top-3 (redundantly in every thread; all identical) ----
    float av[M_SL];
    {
      const float tl   = temp_logit[g];
      const float base = 0.6f / (1.f + __expf(-tl)) + 0.2f;  // sigmoid*0.6+0.2
      const float temp = base / (1.f + shift) + 1e-8f;
      const float invt = 1.f / temp;
#pragma unroll
      for (int m = 0; m < M_SL; ++m) av[m] = att_final[m] * invt;
    }
    float mx = av[0];
#pragma unroll
    for (int m = 1; m < M_SL; ++m) mx = fmaxf(mx, av[m]);
    float sum = 0.f;
#pragma unroll
    for (int m = 0; m < M_SL; ++m) sum += __expf(av[m] - mx);
    const float inv_sum = 1.f / sum;

    int i0 = 0;
    float b0v = av[0];
#pragma unroll
    for (int m = 1; m < M_SL; ++m)
      if (av[m] > b0v) { b0v = av[m]; i0 = m; }
    int i1 = -1;
    float b1v = -3.402823e38f;
#pragma unroll
    for (int m = 0; m < M_SL; ++m)
      if (m != i0 && av[m] > b1v) { b1v = av[m]; i1 = m; }
    int i2 = -1;
    float b2v = -3.402823e38f;
#pragma unroll
    for (int m = 0; m < M_SL; ++m)
      if (m != i0 && m != i1 && av[m] > b2v) { b2v = av[m]; i2 = m; }

    const float v0 = __expf(b0v - mx) * inv_sum;
    const float v1 = __expf(b1v - mx) * inv_sum;
    const float v2 = __expf(b2v - mx) * inv_sum;

    // ---- retrieved[d] = sum_k vals[k]*W[d][idx_k], gathered from LDS ----
#pragma unroll
    for (int r = 0; r < 2; ++r) {
      const int d = tid + r * BLOCK;
      const float* row = wb + d * M_SL;
      out_ret[(size_t)g * D_DIM + d] =
          v0 * row[i0] + v1 * row[i1] + v2 * row[i2];
    }
    if (tid < TOPK) {
      const int   ii = (tid == 0) ? i0 : (tid == 1) ? i1 : i2;
      const float vv = (tid == 0) ? v0 : (tid == 1) ? v1 : v2;
      out_idx[(size_t)g * TOPK + tid] = (float)ii;
      out_val[(size_t)g * TOPK + tid] = vv;
    }
    __syncthreads();  // buf[cur] free for the next iteration's prefetch
  }
}

extern "C" void kernel_launch(void* const* d_in, const int* in_sizes, int n_in,
                              void* d_out, int out_size, void* d_ws,
                              size_t ws_size, hipStream_t stream) {
  const float* q  = (const float*)d_in[0];
  const float* W  = (const float*)d_in[1];
  const float* tl = (const float*)d_in[2];
  const float* sm = (const float*)d_in[3];
  float* out = (float*)d_out;
  hipLaunchKernelGGL(cgm_retrieve_kernel, dim3(NBLK), dim3(BLOCK), 0, stream,
                     q, W, tl, sm, out);
}